// OptimizedSeriesDecomp_9113920602749
// MI455X (gfx1250) — compile-verified
//
#include <hip/hip_runtime.h>

// Problem constants (from reference: x = [16, 4096, 512] f32, kernels 13/17/25)
#define BB 16
#define LL 4096
#define CC 512
#define TL 32            // time-steps per block tile
#define HALO 12          // (25-1)/2, largest half-window
#define ROWS (TL + 2*HALO)   // 56 staged rows
#define NTHREADS (CC / 4)    // 128 threads, one float4 of channels each
#define ROW_BYTES (CC * 4)   // 2048 B per time-step row

typedef float __attribute__((ext_vector_type(4))) vfloat4;
typedef int   __attribute__((ext_vector_type(4))) v4i;

// Address-space-qualified pointer types for the async DMA builtin
// (param 0 is '__attribute__((vector_size(16))) int __device__ *' per hipcc).
typedef __attribute__((address_space(1))) v4i* gas_v4i_ptr;
typedef __attribute__((address_space(3))) v4i* lds_v4i_ptr;

#if __has_builtin(__builtin_amdgcn_global_load_async_to_lds_b128)
#define USE_ASYNC_LDS 1
#endif

__device__ __forceinline__ void stage_one(const float* src, float* dst) {
#ifdef USE_ASYNC_LDS
  __builtin_amdgcn_global_load_async_to_lds_b128(
      (gas_v4i_ptr)src, (lds_v4i_ptr)dst, 0, 0);
#else
  *(vfloat4*)dst = *(const vfloat4*)src;
#endif
}

// Interior-tile staging: one base address, 56 async B128 DMAs with immediate
// offsets (INST_OFFSET advances BOTH the global and the LDS address by the
// same amount -> zero per-row VALU address math). Unrolled via if-constexpr
// so the offset is an ICE.
template <int R>
__device__ __forceinline__ void stage_interior(const float* src, float* dst) {
  if constexpr (R < ROWS) {
#ifdef USE_ASYNC_LDS
    __builtin_amdgcn_global_load_async_to_lds_b128(
        (gas_v4i_ptr)src, (lds_v4i_ptr)dst, R * ROW_BYTES, 0);
#else
    *(vfloat4*)(dst + (size_t)R * CC) = *(const vfloat4*)(src + (size_t)R * CC);
#endif
    stage_interior<R + 1>(src, dst);
  }
}

__device__ __forceinline__ void wait_async0() {
#if __has_builtin(__builtin_amdgcn_s_wait_asynccnt)
  __builtin_amdgcn_s_wait_asynccnt(0);
#else
  asm volatile("s_wait_asynccnt 0" ::: "memory");
#endif
}

__global__ __launch_bounds__(NTHREADS) void
series_decomp_kernel(const float* __restrict__ x,
                     float* __restrict__ seasonal,
                     float* __restrict__ trend) {
  extern __shared__ float smem[];  // ROWS * CC floats = 112 KB

  const int tid  = threadIdx.x;                 // 0..127
  const int tile = blockIdx.x & ((LL / TL) - 1);
  const int b    = blockIdx.x / (LL / TL);
  const int l0   = tile * TL;
  const int coff = tid * 4;                     // channel offset (floats)

  const float* xb = x + (size_t)b * LL * CC;
  float* dst0 = smem + coff;

  // ---- Stage ROWS rows into LDS (single HBM read per element) ----
  if (l0 - HALO >= 0 && l0 + TL + HALO - 1 <= LL - 1) {
    // interior: straight DMA burst, immediate offsets
    const float* src0 = xb + (size_t)(l0 - HALO) * CC + coff;
    stage_interior<0>(src0, dst0);
  } else {
    // boundary: replicate-pad via clamped row index
    for (int r = 0; r < ROWS; ++r) {
      int l = l0 - HALO + r;
      l = (l < 0) ? 0 : ((l > LL - 1) ? LL - 1 : l);
      stage_one(xb + (size_t)l * CC + coff, smem + r * CC + coff);
    }
  }
#ifdef USE_ASYNC_LDS
  wait_async0();
#endif
  __syncthreads();

  // ---- Sliding-window running sums over the staged tile ----
  const float* col = smem + coff;  // row r lives at col + r*CC

#define ROW(r) (*(const vfloat4*)(col + (r) * CC))

  vfloat4 s13 = (vfloat4)0.f, s17 = (vfloat4)0.f, s25 = (vfloat4)0.f;
  // windows for output l0: LDS rows centered at HALO
  for (int r = HALO - 6;  r <= HALO + 6;  ++r) s13 += ROW(r);
  for (int r = HALO - 8;  r <= HALO + 8;  ++r) s17 += ROW(r);
  for (int r = HALO - 12; r <= HALO + 12; ++r) s25 += ROW(r);

  const float c13 = 1.0f / (13.0f * 3.0f);
  const float c17 = 1.0f / (17.0f * 3.0f);
  const float c25 = 1.0f / (25.0f * 3.0f);

  const size_t outBase = ((size_t)b * LL + l0) * CC + coff;
  float* sp = seasonal + outBase;
  float* tp = trend + outBase;

  for (int i = 0; i < TL; ++i) {
    const int cr = i + HALO;            // center row for output l0+i
    vfloat4 cen = ROW(cr);
    vfloat4 t   = s13 * c13 + s17 * c17 + s25 * c25;
    vfloat4 sea = cen - t;

    // streamed outputs: non-temporal B128 stores (don't thrash 192MB L2)
    __builtin_nontemporal_store(sea, (vfloat4*)(sp + (size_t)i * CC));
    __builtin_nontemporal_store(t,   (vfloat4*)(tp + (size_t)i * CC));

    if (i < TL - 1) {   // slide windows; max row touched = TL-2+25 = ROWS-1
      s13 += ROW(cr + 7)  - ROW(cr - 6);
      s17 += ROW(cr + 9)  - ROW(cr - 8);
      s25 += ROW(cr + 13) - ROW(cr - 12);
    }
  }
#undef ROW
}

extern "C" void kernel_launch(void* const* d_in, const int* in_sizes, int n_in,
                              void* d_out, int out_size, void* d_ws, size_t ws_size,
                              hipStream_t stream) {
  const float* x = (const float*)d_in[0];
  float* seasonal = (float*)d_out;
  float* trend    = seasonal + (size_t)BB * LL * CC;

  dim3 grid(BB * (LL / TL));   // 16 * 128 = 2048 blocks
  dim3 block(NTHREADS);        // 128 threads = 4 waves (wave32)
  size_t shmem = (size_t)ROWS * CC * sizeof(float);  // 112 KB dynamic LDS

  series_decomp_kernel<<<grid, block, shmem, stream>>>(x, seasonal, trend);
}